// PillarFeatureNet_32650341384659
// MI455X (gfx1250) — compile-verified
//
#include <hip/hip_runtime.h>
#include <hip/hip_bf16.h>

// ---------------------------------------------------------------------------
// PillarFeatureNet fused pipeline for MI455X (gfx1250, wave32)
//
// x:[4,4,12000,100] f32, indices:[4,12000,2] i32, W:[64,4] f32, gamma/beta:[64]
// out:[4,64,496,432] f32
//
// BN stats computed algebraically from 4x4 second-moment matrix of x
// (y = Wx is linear), so x is read only twice and y is never materialized.
// GEMM uses V_WMMA_F32_16X16X4_F32 (exact K=4 match).
// ---------------------------------------------------------------------------

#define BATCH 4
#define DIM   4
#define NP    12000
#define NN    100
#define NC    64
#define GH    496
#define GW    432
#define NTOT  4800000.0f   // B*P*N
#define NEGBIG (-3.4e38f)

typedef float v2f __attribute__((ext_vector_type(2)));
typedef float v8f __attribute__((ext_vector_type(8)));

// ws float layout: [0..3]=sum_d, [4..19]=second moments (10 used, upper tri),
// [32..95]=a[c] (scale), [96..159]=b[c] (shift)

// ---------------- kernel 1: zero output grid + ws header -------------------
__global__ void zero_kernel(float* __restrict__ out, long n, float* __restrict__ wsf) {
    long stride = (long)gridDim.x * blockDim.x;
    for (long i = (long)blockIdx.x * blockDim.x + threadIdx.x; i < n; i += stride)
        out[i] = 0.0f;
    long t = (long)blockIdx.x * blockDim.x + threadIdx.x;
    if (t < 32) wsf[t] = 0.0f;
}

// ---------------- kernel 2: first/second moments of x over (b,p,n) ---------
__global__ __launch_bounds__(256) void stats_kernel(const float* __restrict__ x,
                                                    float* __restrict__ wsf) {
    const long PN = (long)NP * NN;
    const long total = (long)BATCH * PN;
    float acc[14];
#pragma unroll
    for (int i = 0; i < 14; ++i) acc[i] = 0.0f;

    long stride = (long)gridDim.x * blockDim.x;
    for (long j = (long)blockIdx.x * blockDim.x + threadIdx.x; j < total; j += stride) {
        long b  = j / PN;
        long pn = j - b * PN;
        const float* px = x + (b * DIM) * PN + pn;
        float v0 = px[0];
        float v1 = px[PN];
        float v2 = px[2 * PN];
        float v3 = px[3 * PN];
        acc[0] += v0; acc[1] += v1; acc[2] += v2; acc[3] += v3;
        acc[4]  += v0 * v0; acc[5]  += v0 * v1; acc[6]  += v0 * v2; acc[7]  += v0 * v3;
        acc[8]  += v1 * v1; acc[9]  += v1 * v2; acc[10] += v1 * v3;
        acc[11] += v2 * v2; acc[12] += v2 * v3;
        acc[13] += v3 * v3;
    }
    // wave32 reduction
#pragma unroll
    for (int m = 16; m >= 1; m >>= 1) {
#pragma unroll
        for (int i = 0; i < 14; ++i) acc[i] += __shfl_xor(acc[i], m, 32);
    }
    __shared__ float red[8][14];
    int wave = threadIdx.x >> 5;
    int lane = threadIdx.x & 31;
    if (lane == 0) {
#pragma unroll
        for (int i = 0; i < 14; ++i) red[wave][i] = acc[i];
    }
    __syncthreads();
    if (threadIdx.x < 14) {
        float s = 0.0f;
#pragma unroll
        for (int w = 0; w < 8; ++w) s += red[w][threadIdx.x];
        atomicAdd(&wsf[threadIdx.x < 4 ? threadIdx.x : (4 + threadIdx.x - 4)], s);
    }
}

// ---------------- kernel 3: fold stats + gamma/beta into per-channel a,b ---
__global__ void affine_kernel(const float* __restrict__ Wm,
                              const float* __restrict__ gamma,
                              const float* __restrict__ beta,
                              float* __restrict__ wsf) {
    int c = threadIdx.x;
    if (c >= NC) return;
    float w[4];
#pragma unroll
    for (int d = 0; d < 4; ++d) w[d] = Wm[c * 4 + d];
    float s0 = wsf[0], s1 = wsf[1], s2 = wsf[2], s3 = wsf[3];
    // rebuild symmetric 4x4 moment matrix from upper triangle at wsf[4..13]
    float M00 = wsf[4],  M01 = wsf[5],  M02 = wsf[6],  M03 = wsf[7];
    float M11 = wsf[8],  M12 = wsf[9],  M13 = wsf[10];
    float M22 = wsf[11], M23 = wsf[12], M33 = wsf[13];
    const float inv = 1.0f / NTOT;
    float mean = (w[0] * s0 + w[1] * s1 + w[2] * s2 + w[3] * s3) * inv;
    float ey2 =
        w[0] * (w[0] * M00 + w[1] * M01 + w[2] * M02 + w[3] * M03) +
        w[1] * (w[0] * M01 + w[1] * M11 + w[2] * M12 + w[3] * M13) +
        w[2] * (w[0] * M02 + w[1] * M12 + w[2] * M22 + w[3] * M23) +
        w[3] * (w[0] * M03 + w[1] * M13 + w[2] * M23 + w[3] * M33);
    ey2 *= inv;
    float var = ey2 - mean * mean;
    float a = gamma[c] * rsqrtf(var + 1e-5f);
    wsf[32 + c] = a;
    wsf[96 + c] = beta[c] - mean * a;
}

// ---------------- kernel 4: fused WMMA conv + masked max + scatter ---------
// One wave per (b, pillar). 4 channel-tiles of 16; 7 n-tiles of 16 (n<100 masked).
// A (16x4 f32, 2 VGPRs): lanes0-15 M=0..15 K={0,1}, lanes16-31 M=0..15 K={2,3}.
// B (4x16 f32, 2 VGPRs): lanes0-15 N=0..15 K={0,1}, lanes16-31 N=0..15 K={2,3}.
// C/D (16x16 f32, 8 VGPRs): VGPR r holds channel r (lanes0-15) / r+8 (lanes16-31),
// column = lane%16 -> max over N = elementwise vmax + 16-lane shfl tree.
__global__ __launch_bounds__(256) void pillar_main(const float* __restrict__ x,
                                                   const int* __restrict__ indices,
                                                   const float* __restrict__ Wm,
                                                   const float* __restrict__ wsf,
                                                   float* __restrict__ out) {
    const long PN = (long)NP * NN;
    int wave = blockIdx.x * (blockDim.x >> 5) + (threadIdx.x >> 5);
    int lane = threadIdx.x & 31;
    int half = lane >> 4;          // 0: K=0,1 / M,N 0..15 ; 1: K=2,3
    int lm   = lane & 15;

    int b = wave / NP;
    int p = wave - b * NP;

    // A operands: W tiles, one per 16-channel block
    v2f A[4];
#pragma unroll
    for (int cb = 0; cb < 4; ++cb) {
        const float* wp = Wm + (cb * 16 + lm) * 4 + 2 * half;
        A[cb].x = wp[0];
        A[cb].y = wp[1];
    }

    v8f mx[4];
#pragma unroll
    for (int cb = 0; cb < 4; ++cb)
#pragma unroll
        for (int r = 0; r < 8; ++r) mx[cb][r] = NEGBIG;

    // x rows for this lane's K pair: d = 2*half and 2*half+1
    const float* xb = x + ((long)b * DIM + 2 * half) * PN + (long)p * NN;

#pragma unroll
    for (int t = 0; t < 7; ++t) {
        int col = t * 16 + lm;
        int cc  = col < NN ? col : (NN - 1);   // clamp (masked below), no divergence
        v2f Bv;
        Bv.x = xb[cc];
        Bv.y = xb[PN + cc];

        v8f z = {};
        v8f c0 = __builtin_amdgcn_wmma_f32_16x16x4_f32(false, A[0], false, Bv, (short)0, z, false, false);
        v8f c1 = __builtin_amdgcn_wmma_f32_16x16x4_f32(false, A[1], false, Bv, (short)0, z, false, false);
        v8f c2 = __builtin_amdgcn_wmma_f32_16x16x4_f32(false, A[2], false, Bv, (short)0, z, false, false);
        v8f c3 = __builtin_amdgcn_wmma_f32_16x16x4_f32(false, A[3], false, Bv, (short)0, z, false, false);

        bool valid = col < NN;
#pragma unroll
        for (int r = 0; r < 8; ++r) {
            mx[0][r] = fmaxf(mx[0][r], valid ? c0[r] : NEGBIG);
            mx[1][r] = fmaxf(mx[1][r], valid ? c1[r] : NEGBIG);
            mx[2][r] = fmaxf(mx[2][r], valid ? c2[r] : NEGBIG);
            mx[3][r] = fmaxf(mx[3][r], valid ? c3[r] : NEGBIG);
        }
    }

    // reduce max across the 16 columns held in each half-wave
#pragma unroll
    for (int cb = 0; cb < 4; ++cb) {
#pragma unroll
        for (int r = 0; r < 8; ++r) {
            float v = mx[cb][r];
            v = fmaxf(v, __shfl_xor(v, 1, 32));
            v = fmaxf(v, __shfl_xor(v, 2, 32));
            v = fmaxf(v, __shfl_xor(v, 4, 32));
            v = fmaxf(v, __shfl_xor(v, 8, 32));
            mx[cb][r] = v;
        }
    }

    // scatter: affine (monotone, applied post-max) + relu + atomic add
    const int* ip = indices + ((long)b * NP + p) * 2;
    int fidx = ip[0] * GW + ip[1];

    if (lm == 0) {  // lanes 0 (channels r) and 16 (channels r+8)
        const float* av = wsf + 32;
        const float* bv = wsf + 96;
        float* ob = out + (long)b * NC * (GH * GW) + fidx;
#pragma unroll
        for (int cb = 0; cb < 4; ++cb) {
#pragma unroll
            for (int r = 0; r < 8; ++r) {
                int c = cb * 16 + r + 8 * half;
                float val = fmaxf(av[c] * mx[cb][r] + bv[c], 0.0f);
                atomicAdd(ob + (long)c * (GH * GW), val);
            }
        }
    }
}

// ---------------------------------------------------------------------------
extern "C" void kernel_launch(void* const* d_in, const int* in_sizes, int n_in,
                              void* d_out, int out_size, void* d_ws, size_t ws_size,
                              hipStream_t stream) {
    const float* x       = (const float*)d_in[0];
    const int*   indices = (const int*)d_in[1];
    const float* W       = (const float*)d_in[2];
    const float* gamma   = (const float*)d_in[3];
    const float* beta    = (const float*)d_in[4];
    float* out = (float*)d_out;
    float* wsf = (float*)d_ws;

    zero_kernel<<<4096, 256, 0, stream>>>(out, (long)out_size, wsf);
    stats_kernel<<<512, 256, 0, stream>>>(x, wsf);
    affine_kernel<<<1, 64, 0, stream>>>(W, gamma, beta, wsf);

    int waves  = BATCH * NP;                 // 48000 waves, one per (b,p)
    int blocks = waves / 8;                  // 8 waves (256 threads) per block
    pillar_main<<<blocks, 256, 0, stream>>>(x, indices, W, wsf, out);
}